// Embedding_61022895341672
// MI455X (gfx1250) — compile-verified
//
#include <hip/hip_runtime.h>
#include <cstdint>

typedef float v2f __attribute__((ext_vector_type(2)));
typedef float v8f __attribute__((ext_vector_type(8)));
typedef unsigned int u32x4 __attribute__((ext_vector_type(4)));
typedef int i32x4 __attribute__((ext_vector_type(4)));
typedef int i32x8 __attribute__((ext_vector_type(8)));

#define DD      128
#define NREL    8
#define NNODES  50000
#define NEDGES  800000
#define NLAYERS 10
#define FIN     100

#define STRIDE_T  34    // TRANSB LDS row stride: 32 + pad 2 (pad_interval=4, pad_amount=1)
#define STRIDE_N 132    // non-TRANSB LDS row stride: 128 + pad 4 (pad_interval=6, pad_amount=3)

// Issue one TDM 2D tile load (wave-level op; TENSORcnt-tracked).
__device__ __forceinline__ void tdm_load_tile(unsigned lds_addr, const float* gptr,
                                              unsigned tdim0, unsigned tdim1,
                                              unsigned tile0, unsigned tile1,
                                              unsigned long long stride0,
                                              int pad_interval, int pad_amount)
{
    unsigned long long ga = (unsigned long long)(uintptr_t)gptr;
    u32x4 g0;
    g0[0] = 1u;                                                   // count=1, normal mode
    g0[1] = lds_addr;                                             // LDS byte address
    g0[2] = (unsigned)ga;                                         // global_addr[31:0]
    g0[3] = (unsigned)((ga >> 32) & 0x1FFFFFFu) | (2u << 30);     // global_addr[56:32] | type=2

    i32x8 g1;
    g1[0] = (int)((2u << 16) | (1u << 20) |                       // data_size=4B, pad_enable
                  ((unsigned)pad_interval << 22) | ((unsigned)pad_amount << 25));
    g1[1] = (int)((tdim0 & 0xFFFFu) << 16);                       // tensor_dim0[15:0] (atomic addr=0)
    g1[2] = (int)((tdim0 >> 16) | ((tdim1 & 0xFFFFu) << 16));     // tensor_dim0[31:16] | tensor_dim1[15:0]
    g1[3] = (int)((tdim1 >> 16) | (tile0 << 16));                 // tensor_dim1[31:16] | tile_dim0
    g1[4] = (int)(tile1 & 0xFFFFu);                               // tile_dim1 | tile_dim2=0
    g1[5] = (int)(unsigned)(stride0 & 0xFFFFFFFFull);             // tensor_dim0_stride[31:0]
    g1[6] = (int)(unsigned)((stride0 >> 32) & 0xFFFFull);         // stride0[47:32] | stride1[15:0]=0
    g1[7] = 0;                                                    // stride1 hi = 0

    i32x4 g2 = (i32x4)0;
    i32x4 g3 = (i32x4)0;
#if defined(__clang_major__) && __clang_major__ >= 23
    i32x8 gx = (i32x8)0;
    __builtin_amdgcn_tensor_load_to_lds(g0, g1, g2, g3, gx, 0);
#else
    __builtin_amdgcn_tensor_load_to_lds(g0, g1, g2, g3, 0);
#endif
}

// C[M, Dout] = act( cat(A0|A1|A2)[M,K] @ B + bias ), A chunks are 128 wide.
// TRANSB=false: B row-major [K, Dout] (ldb = Dout)   -> x @ Wrel_r, x @ Wloop
// TRANSB=true : B is W[Dout, K] row-major (ldb = K)  -> in @ W^T
// Block: 8 waves x one 16-row tile; one 128-col group per block (blockIdx.z).
// Weight tile staged into LDS by the Tensor Data Mover (pad -> conflict-free strides).
// Per k-step: one v2f A load + 8 LDS fragment reads + 8 v_wmma_f32_16x16x4_f32.
template <bool TRANSB, bool TANH>
__global__ void __launch_bounds__(256)
wmma_gemm(const float* __restrict__ A0, int lda0,
          const float* __restrict__ A1, int lda1,
          const float* __restrict__ A2, int lda2,
          const float* __restrict__ B, int ldb,
          const float* __restrict__ bias,
          float* __restrict__ C,
          int M, int K, int Dout,
          long strideB, long strideC)
{
    __shared__ float Bs[128 * STRIDE_T];   // 17.4 KB, also covers 32*STRIDE_N = 4224 floats

    const int tid  = threadIdx.x;
    const int wave = tid >> 5;
    const int lane = tid & 31;
    const int half = lane >> 4;     // 0: (K0,K1) pair, 1: (K2,K3) pair
    const int l16  = lane & 15;

    const int  mtiles = M >> 4;
    const int  mtile  = blockIdx.x * 8 + wave;
    const bool active = (mtile < mtiles);   // wave-uniform
    const int  row0   = mtile << 4;
    const int  col0   = blockIdx.z << 7;

    const float* Bb = B + (long)blockIdx.y * strideB;
    float*       Cc = C + (long)blockIdx.y * strideC;

    const unsigned lds_base = (unsigned)(uintptr_t)(&Bs[0]);  // flat->LDS: low 32 bits = LDS offset

    v8f acc[8] = {};

    for (int k0 = 0; k0 < K; k0 += 32) {
        const int kt = (K - k0 < 32) ? (K - k0) : 32;

        __syncthreads();                       // previous tile fully consumed
        if (wave == 0) {
            if (TRANSB) {
                // tile: 32 k (contiguous) x 128 W-rows; OOB k zero-filled (ragged K=100 tail)
                tdm_load_tile(lds_base, Bb + (long)col0 * ldb + k0,
                              (unsigned)(ldb - k0), 128u, 32u, 128u,
                              (unsigned long long)ldb, /*interval 32dw*/4, /*pad 2dw*/1);
            } else {
                // tile: 128 cols (contiguous) x 32 k-rows
                tdm_load_tile(lds_base, Bb + (long)k0 * ldb + col0,
                              128u, (unsigned)kt, 128u, (unsigned)kt,
                              (unsigned long long)ldb, /*interval 128dw*/6, /*pad 4dw*/3);
            }
            __builtin_amdgcn_s_wait_tensorcnt(0);
        }
        __syncthreads();                       // publish tile to all waves

        if (active) {
            // A chunk chosen once per 32-wide k-tile
            const int c = k0 >> 7;
            const float* Ac; int lda;
            if (c == 0)      { Ac = A0; lda = lda0; }
            else if (c == 1) { Ac = A1; lda = lda1; }
            else             { Ac = A2; lda = lda2; }
            const float* Arow = Ac + (long)(row0 + l16) * lda + (k0 & 127) + 2 * half;

            if (kt == 32) {
#pragma unroll
                for (int kk = 0; kk < 32; kk += 4) {
                    v2f a = *(const v2f*)(Arow + kk);
#pragma unroll
                    for (int nt = 0; nt < 8; ++nt) {
                        v2f b;
                        if (TRANSB) {
                            b = *(const v2f*)(&Bs[(nt * 16 + l16) * STRIDE_T + kk + 2 * half]);
                        } else {
                            const int cl = nt * 16 + l16;
                            b[0] = Bs[(kk + 2 * half)     * STRIDE_N + cl];
                            b[1] = Bs[(kk + 2 * half + 1) * STRIDE_N + cl];
                        }
                        acc[nt] = __builtin_amdgcn_wmma_f32_16x16x4_f32(
                            false, a, false, b, (short)0, acc[nt], false, false);
                    }
                }
            } else {
                for (int kk = 0; kk < kt; kk += 4) {
                    v2f a = *(const v2f*)(Arow + kk);
#pragma unroll
                    for (int nt = 0; nt < 8; ++nt) {
                        v2f b;
                        if (TRANSB) {
                            b = *(const v2f*)(&Bs[(nt * 16 + l16) * STRIDE_T + kk + 2 * half]);
                        } else {
                            const int cl = nt * 16 + l16;
                            b[0] = Bs[(kk + 2 * half)     * STRIDE_N + cl];
                            b[1] = Bs[(kk + 2 * half + 1) * STRIDE_N + cl];
                        }
                        acc[nt] = __builtin_amdgcn_wmma_f32_16x16x4_f32(
                            false, a, false, b, (short)0, acc[nt], false, false);
                    }
                }
            }
        }
    }

    if (!active) return;

#pragma unroll
    for (int nt = 0; nt < 8; ++nt) {
        const int col = col0 + nt * 16 + l16;
        const float bv = bias ? bias[col] : 0.0f;
#pragma unroll
        for (int i = 0; i < 8; ++i) {
            const int row = row0 + i + 8 * half;   // C/D layout: VGPR i -> rows i / i+8
            float v = acc[nt][i] + bv;
            if (TANH) v = tanhf(v);
            Cc[(long)row * Dout + col] = v;
        }
    }
}

// msg[dst[e]] += h[etype[e]][src[e]]  (one wave per edge, 4 floats per lane)
__global__ void __launch_bounds__(256)
scatter_edges(const float* __restrict__ h,
              const int* __restrict__ src,
              const int* __restrict__ dst,
              const int* __restrict__ etype,
              float* __restrict__ msg, int E)
{
    int e = blockIdx.x * 8 + (threadIdx.x >> 5);
    if (e >= E) return;
    const int lane = threadIdx.x & 31;
    const float4 v = *(const float4*)(h + (((long)etype[e] * NNODES + src[e]) << 7) + lane * 4);
    float* out = msg + ((long)dst[e] << 7) + lane * 4;
    atomicAdd(out + 0, v.x);
    atomicAdd(out + 1, v.y);
    atomicAdd(out + 2, v.z);
    atomicAdd(out + 3, v.w);
}

extern "C" void kernel_launch(void* const* d_in, const int* in_sizes, int n_in,
                              void* d_out, int out_size, void* d_ws, size_t ws_size,
                              hipStream_t stream)
{
    (void)in_sizes; (void)n_in; (void)out_size; (void)ws_size;
    const float* nodes_fea = (const float*)d_in[0];
    const float* Win   = (const float*)d_in[1];
    const float* bin_  = (const float*)d_in[2];
    const float* Wrel  = (const float*)d_in[3];
    const float* Wloop = (const float*)d_in[4];
    const float* brel  = (const float*)d_in[5];
    const float* W1    = (const float*)d_in[6];
    const float* b1    = (const float*)d_in[7];
    const float* W2    = (const float*)d_in[8];
    const float* b2    = (const float*)d_in[9];
    const int*   src   = (const int*)d_in[10];
    const int*   dst   = (const int*)d_in[11];
    const int*   et    = (const int*)d_in[12];

    float* ws = (float*)d_ws;
    size_t off = 0;
    float* xA  = ws + off; off += (size_t)NNODES * DD;
    float* xB  = ws + off; off += (size_t)NNODES * DD;
    float* msg = ws + off; off += (size_t)NNODES * DD;
    float* mid = ws + off; off += (size_t)NNODES * 2 * DD;
    float* h   = ws + off; off += (size_t)NREL * NNODES * DD;

    const int mtiles = NNODES / 16;           // 3125
    const int blk = (mtiles + 7) / 8;         // 391 blocks of 8 waves

    // x = tanh(nodes_fea @ Win^T + bin)   (K=100; TDM zero-fills the ragged k-tail)
    wmma_gemm<true, true><<<dim3(blk, 1, 1), 256, 0, stream>>>(
        nodes_fea, FIN, nullptr, 0, nullptr, 0,
        Win, FIN, bin_, xA, NNODES, FIN, DD, 0, 0);

    float* x = xA;
    float* xn = xB;
    for (int l = 0; l < NLAYERS; ++l) {
        const float* wr  = Wrel  + (size_t)l * NREL * DD * DD;
        const float* wl  = Wloop + (size_t)l * DD * DD;
        const float* br  = brel  + (size_t)l * DD;
        const float* w1  = W1    + (size_t)l * 2 * DD * 2 * DD;
        const float* bb1 = b1    + (size_t)l * 2 * DD;
        const float* w2  = W2    + (size_t)l * DD * 3 * DD;
        const float* bb2 = b2    + (size_t)l * DD;

        // h[r] = x @ Wrel[r]   (batched over relations via grid.y)
        wmma_gemm<false, false><<<dim3(blk, NREL, 1), 256, 0, stream>>>(
            x, DD, nullptr, 0, nullptr, 0, wr, DD, nullptr, h,
            NNODES, DD, DD, (long)DD * DD, (long)NNODES * DD);

        // msg = x @ Wloop + brel
        wmma_gemm<false, false><<<dim3(blk, 1, 1), 256, 0, stream>>>(
            x, DD, nullptr, 0, nullptr, 0, wl, DD, br, msg,
            NNODES, DD, DD, 0, 0);

        // msg[dst] += h[rel][src]
        scatter_edges<<<(NEDGES + 7) / 8, 256, 0, stream>>>(h, src, dst, et, msg, NEDGES);

        // mid = tanh([x, msg] @ W1^T + b1)   (col groups via grid.z)
        wmma_gemm<true, true><<<dim3(blk, 1, 2), 256, 0, stream>>>(
            x, DD, msg, DD, nullptr, 0, w1, 2 * DD, bb1, mid,
            NNODES, 2 * DD, 2 * DD, 0, 0);

        // x_next = tanh([x, mid] @ W2^T + b2)
        float* out = (l == NLAYERS - 1) ? (float*)d_out : xn;
        wmma_gemm<true, true><<<dim3(blk, 1, 1), 256, 0, stream>>>(
            x, DD, mid, 2 * DD, mid + DD, 2 * DD, w2, 3 * DD, bb2, out,
            NNODES, 3 * DD, DD, 0, 0);

        float* tmp = x; x = xn; xn = tmp;
    }
}